// SystemsOfSprings_89051851915731
// MI455X (gfx1250) — compile-verified
//
#include <hip/hip_runtime.h>

// SystemsOfSprings: one explicit-Euler step for 1,048,576 independent
// 4-mass / 6-edge spring systems.  Memory-bound: 160 B per system round
// trip -> ~7.2 us floor at 23.3 TB/s.  Layout: 4 lanes per system (one
// mass per lane), cross-mass exchange via wave32 quad ds_swizzle.

#define TS_STEP 0.05f
#define L_DIAG  4.2720018727f   // sqrt(4^2 + 1.5^2) = sqrt(18.25)

// Quad-mode ds_swizzle: offset[15]=1, lane i of each quad reads quad-lane
// pattern bits [2i+1:2i].  Register-only lane permute (no LDS memory).
template <int PAT>
__device__ __forceinline__ float qswz(float v) {
    return __int_as_float(__builtin_amdgcn_ds_swizzle(__float_as_int(v), PAT));
}

// Spring force of edge {m, n}:  f += (L/r - 1) * (p_m - p_n)
__device__ __forceinline__ void spring_edge(float px, float py,
                                            float nx, float ny,
                                            float L,
                                            float& fx, float& fy) {
    float dx = px - nx;
    float dy = py - ny;
    float d2 = fmaf(dx, dx, dy * dy);
    float inv_r = __frsqrt_rn(d2);                       // v_rsq_f32
    float c = (d2 > 0.0f) ? fmaf(L, inv_r, -1.0f) : 0.0f; // guard r==0
    fx = fmaf(c, dx, fx);
    fy = fmaf(c, dy, fy);
}

__global__ void __launch_bounds__(256)
springs_step_kernel(const float* __restrict__ x,
                    const float* __restrict__ u,
                    const float* __restrict__ xbar,
                    float* __restrict__ out,
                    int n_sys) {
    const int g = blockIdx.x * 256 + threadIdx.x;   // global lane: sys*4 + q
    const int q = g & 3;                            // mass index within system
    if ((g >> 2) >= n_sys) return;

    // Perfectly coalesced b128 / b64 loads (16 B / 8 B per lane, contiguous).
    const float4 s  = reinterpret_cast<const float4*>(x)[g]; // px,py,vx,vy
    const float2 uv = reinterpret_cast<const float2*>(u)[g]; // ux,uy
    const float  xt = xbar[q * 4 + 0];
    const float  yt = xbar[q * 4 + 1];

    const float px = s.x, py = s.y, vx = s.z, vy = s.w;

    // Gravity + global damping:  r*cos(atan2(y,x)) == x exactly, so
    // Fg = -K_G*(p - pbar) - C_G*v  with K_G=1, C_G=2.
    float fx = -(px - xt) - 2.0f * vx;
    float fy = -(py - yt) - 2.0f * vy;

    // Neighbor positions via quad swizzle (xor 1 / 2 / 3 within the quad).
    const float nx1 = qswz<0x80B1>(px), ny1 = qswz<0x80B1>(py); // q ^ 1
    const float nx2 = qswz<0x804E>(px), ny2 = qswz<0x804E>(py); // q ^ 2
    const float nx3 = qswz<0x801B>(px), ny3 = qswz<0x801B>(py); // q ^ 3

    // Rest length keyed by xor of endpoint indices:
    //   xor 1 -> 4.0   ({0,1},{2,3})
    //   xor 2 -> L9    ({0,2},{1,3})
    //   xor 3 -> 1.5   ({1,2},{3,0})
    // Edge damping vanishes since C_I = 0.
    spring_edge(px, py, nx1, ny1, 4.0f,   fx, fy);
    spring_edge(px, py, nx2, ny2, L_DIAG, fx, fy);
    spring_edge(px, py, nx3, ny3, 1.5f,   fx, fy);

    // Euler update; u feeds the acceleration rows only.
    float4 o;
    o.x = fmaf(vx, TS_STEP, px);
    o.y = fmaf(vy, TS_STEP, py);
    o.z = fmaf(fx + uv.x, TS_STEP, vx);
    o.w = fmaf(fy + uv.y, TS_STEP, vy);
    reinterpret_cast<float4*>(out)[g] = o;   // coalesced b128 store
}

extern "C" void kernel_launch(void* const* d_in, const int* in_sizes, int n_in,
                              void* d_out, int out_size, void* d_ws, size_t ws_size,
                              hipStream_t stream) {
    (void)n_in; (void)out_size; (void)d_ws; (void)ws_size;

    // setup_inputs order: t, x, u, w, xbar
    const float* x    = reinterpret_cast<const float*>(d_in[1]);
    const float* u    = reinterpret_cast<const float*>(d_in[2]);
    const float* xbar = reinterpret_cast<const float*>(d_in[4]);
    float*       out  = reinterpret_cast<float*>(d_out);

    const int n_sys   = in_sizes[1] / 16;     // 1,048,576
    const int threads = n_sys * 4;            // 4 lanes per system
    const int block   = 256;                  // 8 wave32 per workgroup
    const int grid    = (threads + block - 1) / block;

    hipLaunchKernelGGL(springs_step_kernel, dim3(grid), dim3(block), 0, stream,
                       x, u, xbar, out, n_sys);
}